// SheafConvLayer_3246995276434
// MI455X (gfx1250) — compile-verified
//
#include <hip/hip_runtime.h>
#include <hip/hip_bf16.h>

// Sheaf conv layer, MI455X / gfx1250, wave32.
// Phase 1: s = x @ W_r + b_r           (WMMA f32 16x16x4, K-chained 64x)
// Phase 2: edge scatter (atomics in L2; stalk array is 1.6MB -> L2 resident)
// Phase 3: out = gelu(acc @ W_o + b_o) (WMMA f32 16x16x4, 16 column tiles)

#define N_NODES 100000
#define IN_CH   256
#define OUT_CH  256
#define STALK   4
#define N_EDGES 3200000
#define TILES   (N_NODES / 16)   // 6250, exact
#define WAVES_PER_BLOCK 8

typedef __attribute__((ext_vector_type(2))) float v2f;
typedef __attribute__((ext_vector_type(8))) float v8f;

// ---------------- Phase 1: stalk projection + zero accumulator ----------------
__global__ __launch_bounds__(256)
void stalk_proj_kernel(const float* __restrict__ x,
                       const float* __restrict__ W_r,   // (256,4) row-major
                       const float* __restrict__ b_r,   // (4)
                       float* __restrict__ s,           // (N,4)
                       float* __restrict__ acc)         // (N,4) zeroed here
{
    __shared__ float lwr[IN_CH * STALK];   // 4 KB
    __shared__ float lbr[STALK];

    for (int i = threadIdx.x; i < IN_CH * STALK; i += 256) lwr[i] = W_r[i];
    if (threadIdx.x < STALK) lbr[threadIdx.x] = b_r[threadIdx.x];
    __syncthreads();

    const int wave = threadIdx.x >> 5;
    const int lane = threadIdx.x & 31;
    const int tile = blockIdx.x * WAVES_PER_BLOCK + wave;
    if (tile >= TILES) return;                 // whole-wave exit: EXEC stays full

    const int rowbase = tile * 16;
    const int m15 = lane & 15;                 // A: M row / B,D: N column
    const int hi  = lane >> 4;                 // K half-select
    const int nn  = m15 & 3;
    const float wmask = (m15 < STALK) ? 1.0f : 0.0f;   // branchless B zero-pad

    // A: lane reads x[rowbase+m15][kk*4 + 2*hi + {0,1}]
    const float* xrow = x + (size_t)(rowbase + m15) * IN_CH + 2 * hi;

    v8f c = {};
    #pragma unroll 4
    for (int kk = 0; kk < IN_CH / 4; ++kk) {
        float2 av = *reinterpret_cast<const float2*>(xrow + kk * 4);
        v2f a; a[0] = av.x; a[1] = av.y;
        // B VGPR r <- W_r[kk*4 + r + 2*hi][n], n>=4 columns forced to zero
        const int k0 = kk * 4 + 2 * hi;
        v2f b;
        b[0] = lwr[(k0 + 0) * STALK + nn] * wmask;
        b[1] = lwr[(k0 + 1) * STALK + nn] * wmask;
        c = __builtin_amdgcn_wmma_f32_16x16x4_f32(false, a, false, b,
                                                  (short)0, c, false, false);
    }

    // D layout: VGPR r -> M = r + 8*hi, N = m15. Only N<4 is real output.
    if (m15 < STALK) {
        const float bias = lbr[m15];
        #pragma unroll
        for (int r = 0; r < 8; ++r) {
            const int row = rowbase + hi * 8 + r;
            s[row * STALK + m15]   = c[r] + bias;
            acc[row * STALK + m15] = 0.0f;       // fold accumulator clear in
        }
    }
}

// ---------------- Phase 2: edge scatter (diff into acc) ----------------
__global__ __launch_bounds__(256)
void edge_scatter_kernel(const long long* __restrict__ ei,  // (2,E) int64
                         const float* __restrict__ s,       // (N,4)
                         float* __restrict__ acc)           // (N,4)
{
    const int i = blockIdx.x * blockDim.x + threadIdx.x;
    if (i >= N_EDGES) return;
    const int src = (int)ei[i];
    const int dst = (int)ei[N_EDGES + i];

    const float4 sd = *reinterpret_cast<const float4*>(s + (size_t)dst * STALK);
    const float4 ss = *reinterpret_cast<const float4*>(s + (size_t)src * STALK);
    const float d0 = sd.x - ss.x, d1 = sd.y - ss.y;
    const float d2 = sd.z - ss.z, d3 = sd.w - ss.w;

    float* ad = acc + (size_t)dst * STALK;
    float* as = acc + (size_t)src * STALK;
    atomicAdd(ad + 0,  d0); atomicAdd(ad + 1,  d1);
    atomicAdd(ad + 2,  d2); atomicAdd(ad + 3,  d3);
    atomicAdd(as + 0, -d0); atomicAdd(as + 1, -d1);
    atomicAdd(as + 2, -d2); atomicAdd(as + 3, -d3);
}

// ---------------- Phase 3: output projection + exact GELU ----------------
__global__ __launch_bounds__(256)
void out_proj_kernel(const float* __restrict__ acc,  // (N,4)
                     const float* __restrict__ W_o,  // (4,256) row-major
                     const float* __restrict__ b_o,  // (256)
                     float* __restrict__ out)        // (N,256)
{
    __shared__ float lwo[STALK * OUT_CH];  // 4 KB
    __shared__ float lbo[OUT_CH];          // 1 KB

    for (int i = threadIdx.x; i < STALK * OUT_CH; i += 256) lwo[i] = W_o[i];
    for (int i = threadIdx.x; i < OUT_CH; i += 256) lbo[i] = b_o[i];
    __syncthreads();

    const int wave = threadIdx.x >> 5;
    const int lane = threadIdx.x & 31;
    const int tile = blockIdx.x * WAVES_PER_BLOCK + wave;
    if (tile >= TILES) return;

    const int rowbase = tile * 16;
    const int m15 = lane & 15;
    const int hi  = lane >> 4;

    // A (constant across all 16 column tiles): acc[rowbase+m15][2*hi + {0,1}]
    float2 av = *reinterpret_cast<const float2*>(
        acc + (size_t)(rowbase + m15) * STALK + 2 * hi);
    v2f a; a[0] = av.x; a[1] = av.y;

    #pragma unroll 2
    for (int t = 0; t < OUT_CH / 16; ++t) {
        const int col = t * 16 + m15;
        const int k0  = 2 * hi;
        v2f b;
        b[0] = lwo[(k0 + 0) * OUT_CH + col];
        b[1] = lwo[(k0 + 1) * OUT_CH + col];
        v8f z = {};
        v8f d = __builtin_amdgcn_wmma_f32_16x16x4_f32(false, a, false, b,
                                                      (short)0, z, false, false);
        const float bias = lbo[col];
        #pragma unroll
        for (int r = 0; r < 8; ++r) {
            const float v = d[r] + bias;
            const float g = 0.5f * v * (1.0f + erff(v * 0.70710678118654752f));
            out[(size_t)(rowbase + hi * 8 + r) * OUT_CH + col] = g;
        }
    }
}

// ---------------- Launch ----------------
extern "C" void kernel_launch(void* const* d_in, const int* in_sizes, int n_in,
                              void* d_out, int out_size, void* d_ws, size_t ws_size,
                              hipStream_t stream) {
    const float*     x   = (const float*)d_in[0];
    const long long* ei  = (const long long*)d_in[1];   // int64 (2,E)
    const float*     W_r = (const float*)d_in[2];
    const float*     b_r = (const float*)d_in[3];
    const float*     W_o = (const float*)d_in[4];
    const float*     b_o = (const float*)d_in[5];
    float* out = (float*)d_out;

    float* s   = (float*)d_ws;              // (N,4) stalk features, 1.6 MB
    float* acc = s + (size_t)N_NODES * STALK; // (N,4) edge accumulator, 1.6 MB

    const int tile_blocks = (TILES + WAVES_PER_BLOCK - 1) / WAVES_PER_BLOCK; // 782
    const int edge_blocks = (N_EDGES + 255) / 256;                           // 12500

    stalk_proj_kernel<<<tile_blocks, 256, 0, stream>>>(x, W_r, b_r, s, acc);
    edge_scatter_kernel<<<edge_blocks, 256, 0, stream>>>(ei, s, acc);
    out_proj_kernel<<<tile_blocks, 256, 0, stream>>>(acc, W_o, b_o, out);
}